// Lstmcell_29987461660989
// MI455X (gfx1250) — compile-verified
//
#include <hip/hip_runtime.h>

typedef float v2f __attribute__((ext_vector_type(2)));
typedef float v8f __attribute__((ext_vector_type(8)));

#define IN_DIM     256
#define H_DIM      256
#define K_CAT      512
#define M_TILE     64
#define LDS_STRIDE 516   // 512 + 4 pad floats
#define THREADS    256   // 8 wave32 waves

// GRU cell, fully fused, f32 WMMA (V_WMMA_F32_16X16X4_F32).
// Block: 64 batch rows x 256 output cols.
// Wave tiling 1(M) x 8(N): each wave owns ALL 4 M-tiles (64 rows) and
// 2 N-tiles (32 cols). B fragments are reused across the 4 M-tiles,
// giving 8 global loads per 16 WMMAs in the dual-gate loop.
__global__ __launch_bounds__(THREADS) void gru_wmma_f32(
    const float* __restrict__ x,   const float* __restrict__ hp,
    const float* __restrict__ wr,  const float* __restrict__ wz,
    const float* __restrict__ whh, const float* __restrict__ whx,
    const float* __restrict__ br,  const float* __restrict__ bz,
    const float* __restrict__ bh,  float* __restrict__ out)
{
    __shared__ float xc[M_TILE * LDS_STRIDE];   // [x | h] tile, later [x | r*h]

    const int tid  = threadIdx.x;
    const int lane = tid & 31;
    const int wn   = tid >> 5;                  // 0..7 : 32-col N strip per wave
    const long block_row = (long)blockIdx.x * M_TILE;

    // ---- Stage x | h_prev into LDS (row: cols [0,256)=x, [256,512)=h) ----
    for (int i = tid; i < M_TILE * (K_CAT / 4); i += THREADS) {
        const int row = i >> 7;                 // 128 float4 per row
        const int col = (i & 127) * 4;
        float4 v;
        if (col < IN_DIM) v = *(const float4*)(x  + (block_row + row) * IN_DIM + col);
        else              v = *(const float4*)(hp + (block_row + row) * H_DIM + (col - IN_DIM));
        *(float4*)(&xc[row * LDS_STRIDE + col]) = v;
    }
    __syncthreads();

    // ISA 16x4 f32 A layout: lanes 0-15 -> K=k..k+1, lanes 16-31 -> K=k+2..k+3
    const int a_lrow = lane & 15;
    const int a_koff = (lane >> 4) * 2;
    // B 4x16 layout (mirror of A): v0 = row k + 2*(lane>=16), v1 = that row + 1
    const int b_col  = wn * 32 + (lane & 15);
    const int b_krow = (lane >> 4) * 2;

    v8f accR[2][4], accZ[2][4];                 // [N-tile][M-tile]
    #pragma unroll
    for (int nt = 0; nt < 2; ++nt)
        #pragma unroll
        for (int mt = 0; mt < 4; ++mt)
            #pragma unroll
            for (int v = 0; v < 8; ++v) { accR[nt][mt][v] = 0.0f; accZ[nt][mt][v] = 0.0f; }

    // ---- r and z gates: xc[64x512] @ wr / wz ----
    for (int k = 0; k < K_CAT; k += 4) {
        v2f a[4];
        #pragma unroll
        for (int mt = 0; mt < 4; ++mt)
            a[mt] = *(const v2f*)(&xc[(mt * 16 + a_lrow) * LDS_STRIDE + k + a_koff]);
        const long wrow = (long)(k + b_krow) * H_DIM + b_col;
        #pragma unroll
        for (int nt = 0; nt < 2; ++nt) {
            const float* pr = wr + wrow + nt * 16;
            v2f rb; rb.x = pr[0]; rb.y = pr[H_DIM];
            const float* pz = wz + wrow + nt * 16;
            v2f zb; zb.x = pz[0]; zb.y = pz[H_DIM];
            #pragma unroll
            for (int mt = 0; mt < 4; ++mt) {
                accR[nt][mt] = __builtin_amdgcn_wmma_f32_16x16x4_f32(
                    false, a[mt], false, rb, (short)0, accR[nt][mt], false, false);
                accZ[nt][mt] = __builtin_amdgcn_wmma_f32_16x16x4_f32(
                    false, a[mt], false, zb, (short)0, accZ[nt][mt], false, false);
            }
        }
    }
    __syncthreads();   // all A-fragment reads of the h half complete

    // ---- r = sigmoid(accR + br); overwrite h half of LDS with r*h ----
    // C layout: VGPR v, lanes 0-15 -> M=v, lanes 16-31 -> M=v+8; N = lane&15
    #pragma unroll
    for (int nt = 0; nt < 2; ++nt) {
        const int col = wn * 32 + nt * 16 + (lane & 15);
        const float brc = br[col];
        #pragma unroll
        for (int mt = 0; mt < 4; ++mt) {
            #pragma unroll
            for (int v = 0; v < 8; ++v) {
                const int row = mt * 16 + (lane >> 4) * 8 + v;
                float r = accR[nt][mt][v] + brc;
                r = 1.0f / (1.0f + __expf(-r));
                const int idx = row * LDS_STRIDE + IN_DIM + col;
                xc[idx] = r * xc[idx];          // element owned by exactly this lane
                accR[nt][mt][v] = 0.0f;         // reuse as G accumulator
            }
        }
    }
    __syncthreads();   // r*h visible to all waves

    // ---- candidate: G = x @ whx + (r*h) @ whh over the same LDS tile ----
    for (int k = 0; k < IN_DIM; k += 4) {
        v2f a[4];
        #pragma unroll
        for (int mt = 0; mt < 4; ++mt)
            a[mt] = *(const v2f*)(&xc[(mt * 16 + a_lrow) * LDS_STRIDE + k + a_koff]);
        const long wrow = (long)(k + b_krow) * H_DIM + b_col;
        #pragma unroll
        for (int nt = 0; nt < 2; ++nt) {
            const float* p = whx + wrow + nt * 16;
            v2f bb; bb.x = p[0]; bb.y = p[H_DIM];
            #pragma unroll
            for (int mt = 0; mt < 4; ++mt)
                accR[nt][mt] = __builtin_amdgcn_wmma_f32_16x16x4_f32(
                    false, a[mt], false, bb, (short)0, accR[nt][mt], false, false);
        }
    }
    for (int k = 0; k < H_DIM; k += 4) {
        v2f a[4];
        #pragma unroll
        for (int mt = 0; mt < 4; ++mt)
            a[mt] = *(const v2f*)(&xc[(mt * 16 + a_lrow) * LDS_STRIDE + IN_DIM + k + a_koff]);
        const long wrow = (long)(k + b_krow) * H_DIM + b_col;
        #pragma unroll
        for (int nt = 0; nt < 2; ++nt) {
            const float* p = whh + wrow + nt * 16;
            v2f bb; bb.x = p[0]; bb.y = p[H_DIM];
            #pragma unroll
            for (int mt = 0; mt < 4; ++mt)
                accR[nt][mt] = __builtin_amdgcn_wmma_f32_16x16x4_f32(
                    false, a[mt], false, bb, (short)0, accR[nt][mt], false, false);
        }
    }

    // ---- epilogue: h = (1-z)*h_prev + z*tanh(G + bh) ----
    #pragma unroll
    for (int nt = 0; nt < 2; ++nt) {
        const int col = wn * 32 + nt * 16 + (lane & 15);
        const float bzc = bz[col];
        const float bhc = bh[col];
        #pragma unroll
        for (int mt = 0; mt < 4; ++mt) {
            #pragma unroll
            for (int v = 0; v < 8; ++v) {
                const long row = block_row + mt * 16 + (lane >> 4) * 8 + v;
                float z = accZ[nt][mt][v] + bzc;
                z = 1.0f / (1.0f + __expf(-z));
                const float g  = tanhf(accR[nt][mt][v] + bhc);
                const float h0 = hp[row * H_DIM + col];
                const float hn = (1.0f - z) * h0 + z * g;
                __builtin_nontemporal_store(hn, out + row * H_DIM + col);
            }
        }
    }
}

extern "C" void kernel_launch(void* const* d_in, const int* in_sizes, int n_in,
                              void* d_out, int out_size, void* d_ws, size_t ws_size,
                              hipStream_t stream) {
    const float* x   = (const float*)d_in[0];
    const float* hp  = (const float*)d_in[1];
    const float* wr  = (const float*)d_in[2];
    const float* wz  = (const float*)d_in[3];
    const float* whh = (const float*)d_in[4];
    const float* whx = (const float*)d_in[5];
    const float* br  = (const float*)d_in[6];
    const float* bz  = (const float*)d_in[7];
    const float* bh  = (const float*)d_in[8];
    float* out = (float*)d_out;

    const int rows = in_sizes[0] / IN_DIM;          // 32768
    dim3 grid(rows / M_TILE), block(THREADS);
    gru_wmma_f32<<<grid, block, 0, stream>>>(x, hp, wr, wz, whh, whx, br, bz, bh, out);
}